// SparseVariationalPooler_2302102471462
// MI455X (gfx1250) — compile-verified
//
#include <hip/hip_runtime.h>
#include <stdint.h>

// Problem constants (from reference): B=32, E=16384, H*W=64, K read from device.
#define B_DIM 32
#define E_DIM 16384
#define PIX   64
#define TPB   256
#define ELEM_PER_THREAD (E_DIM / TPB)   // 64
#define QP    4                          // pixels per workgroup (quad)
#define INHIBIT_CONSTF 0.01f

// Native clang vector types: valid for __builtin_nontemporal_{load,store}.
typedef float    __attribute__((ext_vector_type(4))) fvec4;
typedef unsigned __attribute__((ext_vector_type(4))) uvec4;

// ---------------- Kernel 1: zero delta ----------------
__global__ void zero_delta_kernel(float* __restrict__ delta) {
  int i = blockIdx.x * blockDim.x + threadIdx.x;
  if (i < E_DIM) delta[i] = 0.0f;
}

// ---------------- Kernel 2: segment-sum scatter ----------------
__global__ void scatter_kernel(const float* __restrict__ vals,
                               const int* __restrict__ seg,
                               float* __restrict__ delta, int nnz) {
  int i = blockIdx.x * blockDim.x + threadIdx.x;
  if (i < nnz) {
    int e = seg[i];
    if (e >= 0 && e < E_DIM)
      atomicAdd(&delta[e], INHIBIT_CONSTF * vals[i]);
  }
}

// ---------------- Kernel 3: 4 pixels per WG, exact k-th largest via LDS radix select ----
// Stage 4 pixel columns (16 contiguous bytes per e-row) into LDS with CDNA5
// async b128 loads; keys live interleaved as keys[e][q]. Then 4x 8-bit radix
// select passes run 4 independent selections simultaneously.
__global__ __launch_bounds__(TPB) void
kwta_threshold_kernel(const float* __restrict__ tensor,
                      const float* __restrict__ boost,
                      const float* __restrict__ bp,
                      const float* __restrict__ delta,
                      const int* __restrict__ kptr,
                      float* __restrict__ thr) {
  __shared__ unsigned keys[E_DIM * QP];      // 256 KB, layout keys[e*4 + q]
  __shared__ unsigned hist[QP * 256];        // 4 KB
  __shared__ unsigned scan[QP * 256];        // 4 KB
  __shared__ unsigned sel_prefix[QP];
  __shared__ unsigned sel_k[QP];

  const int grp = blockIdx.x;                // 0 .. B*16-1
  const int b   = grp >> 4;
  const int p0  = (grp & 15) * QP;           // first pixel of the quad
  const int tid = threadIdx.x;

  const float* tbase = tensor + ((size_t)b * E_DIM) * PIX + p0;  // uniform, 16B aligned
  const float* bbase = boost  + ((size_t)b * E_DIM) * PIX + p0;
  const float  bpv   = bp[0];

  // LDS byte offset of keys[] (flat-address low 32 bits == LDS offset).
  const unsigned keys_base = (unsigned)(uintptr_t)(void*)keys;

  // ---- Stage raw tensor quads into LDS via async b128 loads (GVS mode) ----
  for (int j = 0; j < ELEM_PER_THREAD; ++j) {
    const int e = tid + j * TPB;
    const unsigned voff = (unsigned)e * (PIX * 4u);    // e-row stride: 256 bytes
    const unsigned ldsa = keys_base + (unsigned)e * 16u;
    asm volatile("global_load_async_to_lds_b128 %0, %1, %2"
                 :: "v"(ldsa), "v"(voff), "s"(tbase) : "memory");
  }
  asm volatile("s_wait_asynccnt 0" ::: "memory");

  // ---- Convert in place to monotone keys: v = raw + boost*bp + delta[e] ----
  for (int j = 0; j < ELEM_PER_THREAD; ++j) {
    const int e = tid + j * TPB;
    const uvec4 raw = *(const uvec4*)&keys[e * QP];                       // ds_load_b128
    const fvec4 bo  = __builtin_nontemporal_load((const fvec4*)&bbase[(size_t)e * PIX]);
    const float d   = delta[e];
    uvec4 kv;
    #pragma unroll
    for (int q = 0; q < QP; ++q) {
      const float v = __uint_as_float(raw[q]) + bo[q] * bpv + d;
      const unsigned bits = __float_as_uint(v);
      kv[q] = (bits & 0x80000000u) ? ~bits : (bits | 0x80000000u);
    }
    *(uvec4*)&keys[e * QP] = kv;                                          // ds_store_b128
  }
  const unsigned k0 = (unsigned)kptr[0];
  if (tid < QP) { sel_prefix[tid] = 0u; sel_k[tid] = k0; }
  __syncthreads();

  // ---- 4-pass radix select (MSB digit first), 4 pixels concurrently ----
  for (int d = 3; d >= 0; --d) {
    const unsigned shift = 8u * (unsigned)d;
    const unsigned pmask = (d == 3) ? 0u : (0xFFFFFFFFu << (shift + 8u));
    unsigned pfx[QP], kk[QP];
    #pragma unroll
    for (int q = 0; q < QP; ++q) { pfx[q] = sel_prefix[q]; kk[q] = sel_k[q]; }
    #pragma unroll
    for (int q = 0; q < QP; ++q) hist[q * 256 + tid] = 0u;
    __syncthreads();

    for (int j = 0; j < ELEM_PER_THREAD; ++j) {
      const uvec4 kv = *(const uvec4*)&keys[(tid + j * TPB) * QP];
      #pragma unroll
      for (int q = 0; q < QP; ++q) {
        if ((kv[q] & pmask) == pfx[q])
          atomicAdd(&hist[q * 256 + ((kv[q] >> shift) & 255u)], 1u);
      }
    }
    __syncthreads();

    // Parallel suffix sums: scan[q][t] = sum(hist[q][t..255]), all 4 at once.
    #pragma unroll
    for (int q = 0; q < QP; ++q) scan[q * 256 + tid] = hist[q * 256 + tid];
    __syncthreads();
    #pragma unroll
    for (int off = 1; off < 256; off <<= 1) {
      unsigned v[QP], add[QP];
      #pragma unroll
      for (int q = 0; q < QP; ++q) {
        v[q]   = scan[q * 256 + tid];
        add[q] = (tid + off < 256) ? scan[q * 256 + tid + off] : 0u;
      }
      __syncthreads();
      #pragma unroll
      for (int q = 0; q < QP; ++q) scan[q * 256 + tid] = v[q] + add[q];
      __syncthreads();
    }

    // Unique winner per pixel: above < kk <= incl
    #pragma unroll
    for (int q = 0; q < QP; ++q) {
      const unsigned incl  = scan[q * 256 + tid];
      const unsigned above = (tid < 255) ? scan[q * 256 + tid + 1] : 0u;
      if (above < kk[q] && incl >= kk[q]) {
        sel_prefix[q] = pfx[q] | ((unsigned)tid << shift);
        sel_k[q]      = kk[q] - above;
      }
    }
    __syncthreads();
  }

  if (tid < QP) {
    const unsigned key  = sel_prefix[tid];
    const unsigned bits = (key & 0x80000000u) ? (key & 0x7FFFFFFFu) : ~key;
    thr[b * PIX + p0 + tid] = __uint_as_float(bits);
  }
}

// ---------------- Kernel 4: apply threshold, fully coalesced 16B vectors ----------------
__global__ __launch_bounds__(TPB) void
apply_kernel(const fvec4* __restrict__ t4,
             const fvec4* __restrict__ b4,
             const float* __restrict__ bp,
             const float* __restrict__ delta,
             const fvec4* __restrict__ thr4,
             fvec4* __restrict__ out4, int total4) {
  const int i = blockIdx.x * blockDim.x + threadIdx.x;
  if (i >= total4) return;
  const int pq = i & 15;                // quad-of-pixels within a (b,e) row (64/4)
  const int r  = i >> 4;                // b*E + e
  const int e  = r & (E_DIM - 1);
  const int b  = r >> 14;               // E_DIM == 2^14

  const fvec4 t  = t4[i];                                // RT: L2-resident reuse
  const fvec4 bo = __builtin_nontemporal_load(&b4[i]);   // NT: streamed once
  const float d   = delta[e];
  const float bpv = bp[0];
  const fvec4 th  = thr4[(b << 4) + pq];

  fvec4 o;
  float v;
  v = t.x + bo.x * bpv + d; o.x = (v > 0.f && v >= th.x) ? v : 0.f;
  v = t.y + bo.y * bpv + d; o.y = (v > 0.f && v >= th.y) ? v : 0.f;
  v = t.z + bo.z * bpv + d; o.z = (v > 0.f && v >= th.z) ? v : 0.f;
  v = t.w + bo.w * bpv + d; o.w = (v > 0.f && v >= th.w) ? v : 0.f;
  __builtin_nontemporal_store(o, &out4[i]);              // NT: never re-read
}

// ---------------- Host-side launch ----------------
extern "C" void kernel_launch(void* const* d_in, const int* in_sizes, int n_in,
                              void* d_out, int out_size, void* d_ws, size_t ws_size,
                              hipStream_t stream) {
  (void)n_in; (void)out_size; (void)ws_size;
  const float* tensor = (const float*)d_in[0];   // (B,E,8,8) f32
  const float* boost  = (const float*)d_in[1];   // (B,E,8,8) f32
  const float* vals   = (const float*)d_in[2];   // (NNZ,) f32
  const float* bp     = (const float*)d_in[3];   // (1,) f32
  const int*   idx    = (const int*)d_in[4];     // (2,NNZ) i32
  const int*   kptr   = (const int*)d_in[5];     // scalar k
  const int nnz = in_sizes[2];

  float* delta = (float*)d_ws;                                   // E floats
  float* thr   = (float*)((char*)d_ws + E_DIM * sizeof(float));  // B*64 floats

  zero_delta_kernel<<<(E_DIM + TPB - 1) / TPB, TPB, 0, stream>>>(delta);
  scatter_kernel<<<(nnz + TPB - 1) / TPB, TPB, 0, stream>>>(vals, idx + nnz, delta, nnz);
  kwta_threshold_kernel<<<B_DIM * (PIX / QP), TPB, 0, stream>>>(tensor, boost, bp, delta, kptr, thr);

  const int total4 = B_DIM * E_DIM * (PIX / 4);
  apply_kernel<<<(total4 + TPB - 1) / TPB, TPB, 0, stream>>>(
      (const fvec4*)tensor, (const fvec4*)boost, bp, delta,
      (const fvec4*)thr, (fvec4*)d_out, total4);
}